// LightGCLEncoder_71854802862234
// MI455X (gfx1250) — compile-verified
//
#include <hip/hip_runtime.h>
#include <hip/hip_bf16.h>

// ---- problem constants (match reference) ----
#define USER_COUNT 200000
#define ITEM_COUNT 100000
#define N_NODES    300000
#define EMB        64
#define NNZ_TOTAL  9600000LL
#define BATCH      4096
#define N_LAYERS   3

#define SPMM_CHUNK 512   // nnz per wave

typedef __attribute__((ext_vector_type(2))) float v2f;
typedef __attribute__((ext_vector_type(8))) float v8f;

// Guaranteed-native fp32 global atomic add (no-return form, STOREcnt-tracked,
// round-to-nearest-even in the L2 atomic unit). Avoids any chance of clang
// expanding atomicAdd(float*) into a CAS retry loop.
__device__ __forceinline__ void atom_add_f32(float* p, float v) {
  asm volatile("global_atomic_add_f32 %0, %1, off"
               :
               : "v"(p), "v"(v)
               : "memory");
}

// -------------------------------------------------------------------------
// out[0:BATCH*64)      = user_emb[users[b]][c]   (ego term)
// out[BATCH*64:2B*64)  = item_emb[items[b]][c]
// -------------------------------------------------------------------------
__global__ __launch_bounds__(256) void init_out_kernel(
    const float* __restrict__ user_emb, const float* __restrict__ item_emb,
    const int* __restrict__ users, const int* __restrict__ items,
    float* __restrict__ out) {
  int t = blockIdx.x * blockDim.x + threadIdx.x;   // 0 .. 2*BATCH*EMB-1
  int b = t >> 6;
  int c = t & 63;
  float v;
  if (b < BATCH) {
    v = user_emb[(long long)users[b] * EMB + c];
  } else {
    v = item_emb[(long long)items[b - BATCH] * EMB + c];
  }
  out[t] = v;
}

// -------------------------------------------------------------------------
// SpMM: y[row] += val * x[col] for each nonzero. adj_row is sorted, so each
// wave keeps a register accumulator for its current row and flushes with
// native fp32 atomics only on row changes / chunk boundaries.
// Wave layout: 32 lanes x float2 = 64 emb columns; the per-nnz gather of
// x[col][0:64] is a fully coalesced 256B wave read (L2-resident: x = 76.8MB
// fits in the 192MB L2). Adjacency stream uses non-temporal loads so it
// does not evict x from L2.
// -------------------------------------------------------------------------
__global__ __launch_bounds__(256) void spmm_kernel(
    const int* __restrict__ adj_row, const int* __restrict__ adj_col,
    const float* __restrict__ adj_val, const float* __restrict__ x,
    float* __restrict__ y, long long nnz) {
  const int lane = threadIdx.x & 31;
  const long long wave = ((long long)blockIdx.x * blockDim.x + threadIdx.x) >> 5;
  long long start = wave * SPMM_CHUNK;
  if (start >= nnz) return;                       // wave-uniform exit
  long long end = start + SPMM_CHUNK;
  if (end > nnz) end = nnz;

  const int c2 = lane * 2;                        // this lane's emb columns
  float accx = 0.0f, accy = 0.0f;
  int cur_row = -1;

  for (long long base = start; base < end; base += 32) {
    // cooperative, coalesced, non-temporal load of up to 32 index triples
    long long e = base + lane;
    int r_l = 0, c_l = 0;
    float v_l = 0.0f;
    if (e < end) {
      r_l = __builtin_nontemporal_load(adj_row + e);
      c_l = __builtin_nontemporal_load(adj_col + e);
      v_l = __builtin_nontemporal_load(adj_val + e);
    }
    int n = (int)(end - base);
    if (n > 32) n = 32;
    for (int j = 0; j < n; ++j) {
      int   rj = __shfl(r_l, j, 32);
      int   cj = __shfl(c_l, j, 32);
      float vj = __shfl(v_l, j, 32);
      if (rj != cur_row) {
        if (cur_row >= 0) {
          float* p = y + (long long)cur_row * EMB + c2;
          atom_add_f32(p,     accx);
          atom_add_f32(p + 1, accy);
        }
        accx = 0.0f;
        accy = 0.0f;
        cur_row = rj;
      }
      const float2 xv = *(const float2*)(x + (long long)cj * EMB + c2);
      accx = fmaf(vj, xv.x, accx);
      accy = fmaf(vj, xv.y, accy);
    }
  }
  if (cur_row >= 0) {
    float* p = y + (long long)cur_row * EMB + c2;
    atom_add_f32(p,     accx);
    atom_add_f32(p + 1, accy);
  }
}

// -------------------------------------------------------------------------
// out[b][c] += h[users[b]][c]  /  h[USER_COUNT + items[b]][c]
// -------------------------------------------------------------------------
__global__ __launch_bounds__(256) void gather_add_kernel(
    const float* __restrict__ h,
    const int* __restrict__ users, const int* __restrict__ items,
    float* __restrict__ out) {
  int t = blockIdx.x * blockDim.x + threadIdx.x;
  int b = t >> 6;
  int c = t & 63;
  long long src;
  if (b < BATCH) {
    src = (long long)users[b] * EMB + c;
  } else {
    src = (long long)(USER_COUNT + items[b - BATCH]) * EMB + c;
  }
  out[t] += h[src];
}

// -------------------------------------------------------------------------
// Final scale: out *= 0.25, expressed exactly as fp32 WMMA:
//   D = sum_{kb in {0,4,8,12}} (0.25 * I16[:, kb:kb+4]) x X[kb:kb+4, 0:16]
// One wave per 16x16 tile of the 8192x64 output (2048 tiles = 2048 waves,
// launched exactly so EXEC is all-1s as WMMA requires).
// -------------------------------------------------------------------------
__global__ __launch_bounds__(256) void scale_wmma_kernel(float* __restrict__ out) {
#if __has_builtin(__builtin_amdgcn_wmma_f32_16x16x4_f32)
  const int lane = threadIdx.x & 31;
  const int wave = (blockIdx.x * blockDim.x + threadIdx.x) >> 5;  // 0..2047
  const int tile_row = wave >> 2;   // 0..511  (16-row tiles of 8192 rows)
  const int tile_col = wave & 3;    // 0..3    (16-col tiles of 64 cols)
  const int m  = lane & 15;
  const int hi = lane >> 4;         // lanes 16-31 handle K+2 / M+8 halves
  float* base = out + (long long)tile_row * 16 * EMB + tile_col * 16;

  v8f d = {0.0f, 0.0f, 0.0f, 0.0f, 0.0f, 0.0f, 0.0f, 0.0f};
#pragma unroll
  for (int kb = 0; kb < 16; kb += 4) {
    const int r0 = kb + hi * 2;
    v2f a, b;
    // A (16x4) layout: lanes 0-15 -> K = {kb, kb+1}; lanes 16-31 -> K = {kb+2, kb+3}
    a.x = (m == r0)     ? 0.25f : 0.0f;
    a.y = (m == r0 + 1) ? 0.25f : 0.0f;
    // B (4x16) layout: VGPR0 = row r0, VGPR1 = row r0+1, column = lane&15
    b.x = base[(long long)r0 * EMB + m];
    b.y = base[(long long)(r0 + 1) * EMB + m];
    d = __builtin_amdgcn_wmma_f32_16x16x4_f32(
        /*neg_a=*/false, a, /*neg_b=*/false, b,
        /*c_mod=*/(short)0, d, /*reuse_a=*/false, /*reuse_b=*/false);
  }
  // D (16x16) layout: VGPR v holds row v (lanes 0-15) / row v+8 (lanes 16-31)
#pragma unroll
  for (int v = 0; v < 8; ++v) {
    base[(long long)(v + hi * 8) * EMB + m] = d[v];
  }
#else
  int t = blockIdx.x * blockDim.x + threadIdx.x;
  out[t] *= 0.25f;
#endif
}

// -------------------------------------------------------------------------
extern "C" void kernel_launch(void* const* d_in, const int* in_sizes, int n_in,
                              void* d_out, int out_size, void* d_ws, size_t ws_size,
                              hipStream_t stream) {
  const float* user_emb = (const float*)d_in[0];
  const float* item_emb = (const float*)d_in[1];
  const int*   adj_row  = (const int*)d_in[2];
  const int*   adj_col  = (const int*)d_in[3];
  const float* adj_val  = (const float*)d_in[4];
  const int*   users    = (const int*)d_in[5];
  const int*   items    = (const int*)d_in[6];
  float*       out      = (float*)d_out;

  const size_t hElems = (size_t)N_NODES * EMB;
  float* h0 = (float*)d_ws;
  float* h1 = h0 + hElems;

  // h0 = concat(user_emb, item_emb)
  hipMemcpyAsync(h0, user_emb, sizeof(float) * (size_t)USER_COUNT * EMB,
                 hipMemcpyDeviceToDevice, stream);
  hipMemcpyAsync(h0 + (size_t)USER_COUNT * EMB, item_emb,
                 sizeof(float) * (size_t)ITEM_COUNT * EMB,
                 hipMemcpyDeviceToDevice, stream);

  // out = ego term gathers
  {
    int threads = 2 * BATCH * EMB;                    // 524288
    init_out_kernel<<<threads / 256, 256, 0, stream>>>(user_emb, item_emb,
                                                       users, items, out);
  }

  const long long waves  = (NNZ_TOTAL + SPMM_CHUNK - 1) / SPMM_CHUNK;  // 18750
  const int spmm_blocks  = (int)((waves + 7) / 8);                     // 2344

  float* cur = h0;
  float* nxt = h1;
  for (int layer = 0; layer < N_LAYERS; ++layer) {
    hipMemsetAsync(nxt, 0, sizeof(float) * hElems, stream);
    spmm_kernel<<<spmm_blocks, 256, 0, stream>>>(adj_row, adj_col, adj_val,
                                                 cur, nxt, NNZ_TOTAL);
    gather_add_kernel<<<(2 * BATCH * EMB) / 256, 256, 0, stream>>>(nxt, users,
                                                                   items, out);
    float* t = cur; cur = nxt; nxt = t;
  }

  // out *= 0.25 via fp32 WMMA (2048 tiles -> exactly 2048 waves)
  scale_wmma_kernel<<<256, 256, 0, stream>>>(out);
}